// LaggedEdgeScorer_50697793962508
// MI455X (gfx1250) — compile-verified
//
#include <hip/hip_runtime.h>
#include <hip/hip_bf16.h>
#include <math.h>

typedef float v2f __attribute__((ext_vector_type(2)));
typedef float v8f __attribute__((ext_vector_type(8)));

#define NC   32
#define NT   512
#define ND   64
#define NH   128
#define NK1  192   // 3*D

// LDS layout (in floats)
#define OFF_W1T 0        // 128 * 192
#define OFF_W2T 24576    // 128 * 128
#define OFF_W3  40960    // 128
#define OFF_F   41088    // 32 * 192
#define OFF_X   47232    // 32 * 128
#define OFF_X2  51328    // 32 * 128
#define OFF_P   55424    // 32 * 8
#define SMEM_FLOATS 55680

__device__ __forceinline__ float gelu_exact(float x) {
    return 0.5f * x * (1.0f + erff(x * 0.70710678118654752440f));
}
__device__ __forceinline__ float sigmoidf_(float x) {
    return 1.0f / (1.0f + expf(-x));
}

// Async global->LDS copy of one DWORD per lane (per-lane scattered src/dst),
// tracked by ASYNCcnt.  Lowers to global_load_async_to_lds_b32.
__device__ __forceinline__ void async_copy_b32(const float* __restrict__ g, float* l) {
    auto gp = (__attribute__((address_space(1))) int*)(unsigned long long)g;
    auto lp = (__attribute__((address_space(3))) int*)l;
    __builtin_amdgcn_global_load_async_to_lds_b32(gp, lp, 0, 0);
}
__device__ __forceinline__ void wait_async_zero() {
    asm volatile("s_wait_asynccnt 0" ::: "memory");
}

__global__ __launch_bounds__(256)
void edge_mlp_kernel(const float* __restrict__ h,
                     const float* __restrict__ tau_raw,
                     const float* __restrict__ W1, const float* __restrict__ b1,
                     const float* __restrict__ W2, const float* __restrict__ b2,
                     const float* __restrict__ W3, const float* __restrict__ b3,
                     float* __restrict__ ep, float* __restrict__ tau_out) {
    extern __shared__ float smem[];
    float* sW1t = smem + OFF_W1T;   // [n (128)][k (192)] = W1[k][n]
    float* sW2t = smem + OFF_W2T;   // [n (128)][k (128)] = W2[k][n]
    float* sW3  = smem + OFF_W3;    // [128]
    float* sF   = smem + OFF_F;     // [32][192] feature tile
    float* sX   = smem + OFF_X;     // [32][128] layer-1 activations
    float* sX2  = smem + OFF_X2;    // [32][128] layer-2 activations
    float* sP   = smem + OFF_P;     // [32][8]   layer-3 partials

    const int p = blockIdx.x;       // pair index
    const int i = p >> 5;           // target channel
    const int j = p & 31;           // source channel
    const int tid    = threadIdx.x;
    const int lane   = tid & 31;
    const int wave   = tid >> 5;
    const int lanelo = lane & 15;
    const int lhalf  = lane >> 4;   // 0 or 1
    const int nbase  = wave * 16;   // each wave owns 16 output columns

    // ---- stage weights (transposed) directly into LDS via async copies ----
    for (int idx = tid; idx < NK1 * NH; idx += 256) {
        int k = idx >> 7, n = idx & 127;
        async_copy_b32(W1 + idx, &sW1t[n * NK1 + k]);
    }
    for (int idx = tid; idx < NH * NH; idx += 256) {
        int k = idx >> 7, n = idx & 127;
        async_copy_b32(W2 + idx, &sW2t[n * NH + k]);
    }
    if (tid < NH) async_copy_b32(W3 + tid, &sW3[tid]);

    // ---- per-pair delay parameters (overlaps with async weight staging) ----
    const float traw = tau_raw[p];
    const float tauv = (i == j) ? 0.0f : sigmoidf_(traw);
    if (tid == 0) tau_out[p] = tauv;          // tau output (TAU_MAX=1, masked)
    const float nfl = floorf(tauv);           // == 0 for sigmoid in (0,1)
    const int   dn  = (int)nfl;
    const float fr  = tauv - nfl;

    const float bias1 = b1[nbase + lanelo];
    const float bias2 = b2[nbase + lanelo];
    const float bias3 = b3[0];

    const int r8  = tid >> 3;   // 0..31 feature row
    const int seg = tid & 7;    // 0..7
    const int d0  = seg * 8;

    wait_async_zero();          // each wave drains its own async copies
    __syncthreads();            // then LDS is consistent workgroup-wide

    for (int m0 = 0; m0 < NT; m0 += 32) {
        // ---- (a) build feature tile: [target | delayed | abs diff] ----
        {
            const int t  = m0 + r8;
            int i0 = t - dn;     i0 = i0 < 0 ? 0 : (i0 > NT - 1 ? NT - 1 : i0);
            int i1 = t - dn - 1; i1 = i1 < 0 ? 0 : (i1 > NT - 1 ? NT - 1 : i1);
            const float* hi  = h + ((size_t)(i * NT + t))  * ND + d0;
            const float* hj0 = h + ((size_t)(j * NT + i0)) * ND + d0;
            const float* hj1 = h + ((size_t)(j * NT + i1)) * ND + d0;
            const float omf = 1.0f - fr;
            // warm the next tile's rows while the WMMA loops below run
            if (m0 + 32 < NT) {
                __builtin_prefetch(hi + 32 * ND, 0, 3);
                __builtin_prefetch(hj0 + 32 * ND, 0, 3);
            }
#pragma unroll
            for (int c = 0; c < 8; c += 4) {
                float4 tg = *(const float4*)(hi + c);
                float4 a0 = *(const float4*)(hj0 + c);
                float4 a1 = *(const float4*)(hj1 + c);
                float4 dl, ab;
                dl.x = omf * a0.x + fr * a1.x;  dl.y = omf * a0.y + fr * a1.y;
                dl.z = omf * a0.z + fr * a1.z;  dl.w = omf * a0.w + fr * a1.w;
                ab.x = fabsf(tg.x - dl.x);      ab.y = fabsf(tg.y - dl.y);
                ab.z = fabsf(tg.z - dl.z);      ab.w = fabsf(tg.w - dl.w);
                *(float4*)&sF[r8 * NK1 +       d0 + c] = tg;
                *(float4*)&sF[r8 * NK1 +  ND + d0 + c] = dl;
                *(float4*)&sF[r8 * NK1 + 2*ND + d0 + c] = ab;
            }
        }
        __syncthreads();

        // ---- (b) layer 1: [32 x 192] @ [192 x 128] via f32 WMMA ----
        {
            v8f acc0, acc1;
#pragma unroll
            for (int g = 0; g < 8; ++g) { acc0[g] = bias1; acc1[g] = bias1; }
            const float* bptr = &sW1t[(nbase + lanelo) * NK1];
            const float* a0p  = &sF[lanelo * NK1];
            const float* a1p  = &sF[(16 + lanelo) * NK1];
#pragma unroll 8
            for (int kk = 0; kk < NK1; kk += 4) {
                const int kA = kk + lhalf * 2;
                v2f bf = *(const v2f*)(bptr + kA);
                v2f a0 = *(const v2f*)(a0p + kA);
                v2f a1 = *(const v2f*)(a1p + kA);
                acc0 = __builtin_amdgcn_wmma_f32_16x16x4_f32(false, a0, false, bf,
                                                             (short)0, acc0, false, false);
                acc1 = __builtin_amdgcn_wmma_f32_16x16x4_f32(false, a1, false, bf,
                                                             (short)0, acc1, false, false);
            }
#pragma unroll
            for (int g = 0; g < 8; ++g) {
                const int r = g + lhalf * 8;
                sX[r        * NH + nbase + lanelo] = gelu_exact(acc0[g]);
                sX[(16 + r) * NH + nbase + lanelo] = gelu_exact(acc1[g]);
            }
        }
        __syncthreads();

        // ---- (c) layer 2: [32 x 128] @ [128 x 128] via f32 WMMA ----
        {
            v8f acc0, acc1;
#pragma unroll
            for (int g = 0; g < 8; ++g) { acc0[g] = bias2; acc1[g] = bias2; }
            const float* bptr = &sW2t[(nbase + lanelo) * NH];
            const float* a0p  = &sX[lanelo * NH];
            const float* a1p  = &sX[(16 + lanelo) * NH];
#pragma unroll 8
            for (int kk = 0; kk < NH; kk += 4) {
                const int kA = kk + lhalf * 2;
                v2f bf = *(const v2f*)(bptr + kA);
                v2f a0 = *(const v2f*)(a0p + kA);
                v2f a1 = *(const v2f*)(a1p + kA);
                acc0 = __builtin_amdgcn_wmma_f32_16x16x4_f32(false, a0, false, bf,
                                                             (short)0, acc0, false, false);
                acc1 = __builtin_amdgcn_wmma_f32_16x16x4_f32(false, a1, false, bf,
                                                             (short)0, acc1, false, false);
            }
#pragma unroll
            for (int g = 0; g < 8; ++g) {
                const int r = g + lhalf * 8;
                sX2[r        * NH + nbase + lanelo] = gelu_exact(acc0[g]);
                sX2[(16 + r) * NH + nbase + lanelo] = gelu_exact(acc1[g]);
            }
        }
        __syncthreads();

        // ---- (d) layer 3 partial dot with W3 ----
        {
            float partial = 0.0f;
            const float* xr = &sX2[r8 * NH + seg * 16];
            const float* wr = &sW3[seg * 16];
#pragma unroll
            for (int k = 0; k < 16; ++k) partial += xr[k] * wr[k];
            sP[r8 * 8 + seg] = partial;
        }
        __syncthreads();

        // ---- (e) finalize 32 logits -> edge_probs[t][i][j] ----
        if (tid < 32) {
            float s = bias3;
#pragma unroll
            for (int q = 0; q < 8; ++q) s += sP[tid * 8 + q];
            const float prob = (i == j) ? 0.0f : sigmoidf_(s);
            const int t = m0 + tid;
            ep[((size_t)t * NC + i) * NC + j] = prob;
        }
        __syncthreads();
    }
}

// top-8 per (t, i) row of 32 probs -> adjacency (== soft) and hard
__global__ __launch_bounds__(256)
void topk_kernel(const float* __restrict__ ep,
                 float* __restrict__ adj, float* __restrict__ hard) {
    const int row = blockIdx.x * blockDim.x + threadIdx.x;  // 0 .. T*C-1
    if (row >= NT * NC) return;
    const float* v = ep + (size_t)row * NC;
    unsigned sel = 0u;
#pragma unroll
    for (int k = 0; k < 8; ++k) {
        float best = -1e30f;
        int   bi   = 0;
        for (int jj = 0; jj < NC; ++jj) {
            if ((sel >> jj) & 1u) continue;
            const float x = v[jj];
            if (x > best) { best = x; bi = jj; }   // ties -> lowest index
        }
        sel |= 1u << bi;
    }
    for (int jj = 0; jj < NC; ++jj) {
        const float x = v[jj];
        const bool  s = (sel >> jj) & 1u;
        adj[(size_t)row * NC + jj]  = s ? x : 0.0f;
        hard[(size_t)row * NC + jj] = s ? 1.0f : 0.0f;
    }
}

extern "C" void kernel_launch(void* const* d_in, const int* in_sizes, int n_in,
                              void* d_out, int out_size, void* d_ws, size_t ws_size,
                              hipStream_t stream) {
    (void)in_sizes; (void)n_in; (void)out_size; (void)d_ws; (void)ws_size;
    const float* h       = (const float*)d_in[0];
    const float* tau_raw = (const float*)d_in[1];
    const float* W1 = (const float*)d_in[2];
    const float* b1 = (const float*)d_in[3];
    const float* W2 = (const float*)d_in[4];
    const float* b2 = (const float*)d_in[5];
    const float* W3 = (const float*)d_in[6];
    const float* b3 = (const float*)d_in[7];

    float* out    = (float*)d_out;
    float* ep     = out;                      // (1,512,32,32)
    float* adj    = out + 524288;             // (1,512,32,32)
    float* tauOut = out + 1048576;            // (32,32)
    float* hard   = out + 1049600;            // (1,512,32,32)

    const size_t shmem = (size_t)SMEM_FLOATS * sizeof(float);  // ~218 KB < 320 KB WGP LDS
    (void)hipFuncSetAttribute((const void*)edge_mlp_kernel,
                              hipFuncAttributeMaxDynamicSharedMemorySize, (int)shmem);

    edge_mlp_kernel<<<NC * NC, 256, shmem, stream>>>(h, tau_raw, W1, b1, W2, b2, W3, b3,
                                                     ep, tauOut);
    topk_kernel<<<(NT * NC + 255) / 256, 256, 0, stream>>>(ep, adj, hard);
}